// GaussianSpot_16028817948741
// MI455X (gfx1250) — compile-verified
//
#include <hip/hip_runtime.h>
#include <hip/hip_bf16.h>

typedef __attribute__((ext_vector_type(2))) float v2f;
typedef __attribute__((ext_vector_type(8))) float v8f;

#define DCONST 14
#define FCONST 500
#define INV_TWO_PI 0.15915494309189535f

// One wave32 per spot (k,b). Separable Gaussian => rank-1 outer product via
// V_WMMA_F32_16X16X4_F32 (A col0 = amp*gx, B row0 = gy, other K slots zero).
__global__ __launch_bounds__(256) void gaussian_spot_wmma(
    const float* __restrict__ height,
    const float* __restrict__ width,
    const float* __restrict__ x,
    const float* __restrict__ y,
    const float* __restrict__ target_locs,
    const int* __restrict__ n_idx,
    const int* __restrict__ f_idx,
    float* __restrict__ out,
    int B, int KB)
{
    const int wave = blockIdx.x * (blockDim.x >> 5) + (threadIdx.x >> 5);
    if (wave >= KB) return;                 // uniform per wave: EXEC stays all-1s
    const int lid = threadIdx.x & 31;

    const int b = wave % B;                 // (k = wave / B, flat kb = wave)

    // Per-spot scalars (wave-uniform loads).
    const float h  = height[wave];
    const float w  = width[wave];
    const float xo = x[wave];
    const float yo = y[wave];
    const int   ni = n_idx[b];
    const int   fi = f_idx[b];
    const size_t tbase = ((size_t)ni * FCONST + (size_t)fi) * 2;
    const float sx = target_locs[tbase + 0] + xo;   // row (i) center
    const float sy = target_locs[tbase + 1] + yo;   // col (j) center

    const float w2     = w * w;
    const float inv2w2 = 0.5f / w2;
    const float amp    = h * INV_TWO_PI / w2;       // h / (2*pi*w^2)

    // lane t in [0,16): compute gx[t] (amp folded in) and gy[t].
    const int   t  = lid & 15;
    const float ft = (float)t;
    const float dx = ft - sx;
    const float dy = ft - sy;
    const float gx = amp * __expf(-dx * dx * inv2w2);
    const float gy =       __expf(-dy * dy * inv2w2);

    const bool lo = (lid < 16);
    // A (16x4 f32, 2 VGPRs): lanes 0-15 hold {K=0,K=1}; lanes 16-31 {K=2,K=3}.
    // Only K=0 carries data; all other K slots are zero.
    v2f a;
    a.x = lo ? gx : 0.0f;
    a.y = 0.0f;
    // B (4x16 f32): row K=0 across lanes 0-15 of VGPR0; everything else zero.
    v2f bm;
    bm.x = lo ? gy : 0.0f;
    bm.y = 0.0f;

    v8f c = {};
    // D[m,n] = sum_k A[m,k]*B[k,n] = gx[m]*gy[n]
    c = __builtin_amdgcn_wmma_f32_16x16x4_f32(
            /*neg_a=*/false, a, /*neg_b=*/false, bm,
            /*c_mod=*/(short)0, c, /*reuse_a=*/false, /*reuse_b=*/false);

    // C/D layout: VGPR r -> lanes 0-15: (M=r,   N=lane),
    //                       lanes 16-31:(M=r+8, N=lane-16).
    float* __restrict__ o = out + (size_t)wave * (DCONST * DCONST);
    const int n     = t;
    const int baseM = lo ? 0 : 8;
    if (n < DCONST) {
        #pragma unroll
        for (int r = 0; r < 8; ++r) {
            const int m = baseM + r;
            if (m < DCONST) {
                o[m * DCONST + n] = c[r];
            }
        }
    }
}

extern "C" void kernel_launch(void* const* d_in, const int* in_sizes, int n_in,
                              void* d_out, int out_size, void* d_ws, size_t ws_size,
                              hipStream_t stream) {
    const float* height      = (const float*)d_in[0];
    const float* width       = (const float*)d_in[1];
    const float* x           = (const float*)d_in[2];
    const float* y           = (const float*)d_in[3];
    const float* target_locs = (const float*)d_in[4];
    const int*   n_idx       = (const int*)d_in[5];
    const int*   f_idx       = (const int*)d_in[6];
    float*       out         = (float*)d_out;

    const int KB = in_sizes[0];   // K*B = 200000 spots
    const int B  = in_sizes[5];   // 100000

    const int wavesPerBlock = 8;  // 256 threads, wave32
    const int blocks = (KB + wavesPerBlock - 1) / wavesPerBlock;

    gaussian_spot_wmma<<<blocks, 256, 0, stream>>>(
        height, width, x, y, target_locs, n_idx, f_idx, out, B, KB);
}